// GraphConvolution_15178414424503
// MI455X (gfx1250) — compile-verified
//
#include <hip/hip_runtime.h>

typedef __attribute__((ext_vector_type(2))) float v2f;
typedef __attribute__((ext_vector_type(8))) float v8f;

#define DFEAT 128
#define SPMM_CHUNK 128

// ---------------------------------------------------------------------------
// Kernel: zero-fill (float4 granularity, grid-stride)
// ---------------------------------------------------------------------------
__global__ __launch_bounds__(256) void zero_f32(float* __restrict__ p, int n4) {
  float4 z = make_float4(0.f, 0.f, 0.f, 0.f);
  for (int i = blockIdx.x * blockDim.x + threadIdx.x; i < n4;
       i += gridDim.x * blockDim.x) {
    reinterpret_cast<float4*>(p)[i] = z;
  }
}

// ---------------------------------------------------------------------------
// Kernel: in-place ReLU (float4 granularity, grid-stride)
// ---------------------------------------------------------------------------
__global__ __launch_bounds__(256) void relu_f32(float* __restrict__ p, int n4) {
  for (int i = blockIdx.x * blockDim.x + threadIdx.x; i < n4;
       i += gridDim.x * blockDim.x) {
    float4 v = reinterpret_cast<float4*>(p)[i];
    v.x = fmaxf(v.x, 0.f);
    v.y = fmaxf(v.y, 0.f);
    v.z = fmaxf(v.z, 0.f);
    v.w = fmaxf(v.w, 0.f);
    reinterpret_cast<float4*>(p)[i] = v;
  }
}

// ---------------------------------------------------------------------------
// Kernel: pre = x @ W  using V_WMMA_F32_16X16X4_F32 (full f32 precision).
//
// Block = 256 threads = 8 wave32s. W (128x128 f32 = 64 KB) is cached in LDS;
// each wave owns one 16-column N-tile and preloads its 32 B-fragments
// (K-steps of 4) into registers. The block then grid-strides over 16-row
// M-tiles: stage the 16x128 x-tile in LDS (coalesced float4 loads), then
// each wave runs 32 back-to-back WMMAs into an 8-VGPR f32 accumulator.
//
// VGPR layouts per CDNA5 ISA 7.12.2:
//   A 16x4 f32 : lane<16 holds row M=lane, (K=0,K=1); lane>=16 holds (K=2,K=3)
//   B 4x16 f32 : lane<16 holds col N=lane, (K=0,K=1); lane>=16 holds (K=2,K=3)
//   C/D 16x16  : VGPR r -> M=r (lanes 0-15) / M=r+8 (lanes 16-31), N=lane%16
// ---------------------------------------------------------------------------
__global__ __launch_bounds__(256) void gemm_xw_wmma(
    const float* __restrict__ x, const float* __restrict__ W,
    float* __restrict__ pre, int nrows) {
  __shared__ float Wlds[DFEAT * DFEAT];  // 64 KB
  __shared__ float Xlds[16 * DFEAT];     // 8 KB

  const int tid = threadIdx.x;
  const int wave = tid >> 5;      // 0..7 -> N tile
  const int lane = tid & 31;
  const int half = lane >> 4;     // 0: K={0,1}+4kk, 1: K={2,3}+4kk
  const int l16 = lane & 15;
  const int nb = wave * 16;       // this wave's output column base

  // Stage W into LDS (16384 floats / 256 threads = 16 float4 each, coalesced)
  for (int i = tid * 4; i < DFEAT * DFEAT; i += 256 * 4) {
    *reinterpret_cast<float4*>(&Wlds[i]) =
        *reinterpret_cast<const float4*>(&W[i]);
  }
  __syncthreads();

  // Preload all 32 B fragments for this wave's N-tile into registers (64 VGPRs)
  v2f Bfrag[32];
#pragma unroll
  for (int kk = 0; kk < 32; ++kk) {
    const int k0 = 4 * kk + 2 * half;
    Bfrag[kk][0] = Wlds[k0 * DFEAT + nb + l16];
    Bfrag[kk][1] = Wlds[(k0 + 1) * DFEAT + nb + l16];
  }

  const int ntiles = nrows >> 4;  // N = 100000 -> 6250 exact 16-row tiles
  for (int tile = blockIdx.x; tile < ntiles; tile += gridDim.x) {
    const int row0 = tile << 4;

    __syncthreads();  // protect Xlds from previous iteration's readers
    // Stage 16x128 x-tile: 2048 floats / 256 threads = 8 floats each
    {
      const int base = tid * 8;
      *reinterpret_cast<float4*>(&Xlds[base]) =
          *reinterpret_cast<const float4*>(&x[row0 * DFEAT + base]);
      *reinterpret_cast<float4*>(&Xlds[base + 4]) =
          *reinterpret_cast<const float4*>(&x[row0 * DFEAT + base + 4]);
    }
    __syncthreads();

    v8f acc = {};
#pragma unroll
    for (int kk = 0; kk < 32; ++kk) {
      const int k0 = 4 * kk + 2 * half;
      // A fragment: row M=l16, elements (k0, k0+1) -> contiguous 8B LDS load
      v2f Afrag = *reinterpret_cast<const v2f*>(&Xlds[l16 * DFEAT + k0]);
      acc = __builtin_amdgcn_wmma_f32_16x16x4_f32(
          /*neg_a=*/false, Afrag, /*neg_b=*/false, Bfrag[kk],
          /*c_mod=*/(short)0, acc, /*reuse_a=*/false, /*reuse_b=*/false);
    }

    // Store D: VGPR r -> row (row0 + half*8 + r), col (nb + l16)
    float* outp = pre + (row0 + half * 8) * DFEAT + nb + l16;
#pragma unroll
    for (int r = 0; r < 8; ++r) {
      outp[r * DFEAT] = acc[r];
    }
  }
}

// ---------------------------------------------------------------------------
// Kernel: out[row[e]] += pre[col[e]] * vals[e]  (rows sorted ascending)
//
// One 128-thread block (4 waves) per SPMM_CHUNK edges; thread t owns feature
// dim t. Edge metadata staged in LDS (broadcast reads). Runs of equal
// destination row accumulate in a register; a global_atomic_add_f32 burst is
// emitted only when the row changes or at the chunk boundary (~E/16 bursts).
// pre (51 MB) is L2-resident (192 MB L2) so the per-edge gather of 512 B is
// served from L2, not HBM.
// ---------------------------------------------------------------------------
__global__ __launch_bounds__(SPMM_CHUNK) void spmm_scatter(
    const float* __restrict__ pre, const int* __restrict__ row,
    const int* __restrict__ col, const float* __restrict__ vals,
    float* __restrict__ out, int nE) {
  __shared__ int srow[SPMM_CHUNK];
  __shared__ int scol[SPMM_CHUNK];
  __shared__ float sval[SPMM_CHUNK];

  const int t = threadIdx.x;  // feature dim 0..127
  const int e0 = blockIdx.x * SPMM_CHUNK;
  const int nthis = min(SPMM_CHUNK, nE - e0);
  if (nthis <= 0) return;

  if (t < nthis) {
    srow[t] = row[e0 + t];
    scol[t] = col[e0 + t];
    sval[t] = vals[e0 + t];
  }
  __syncthreads();

  int cur = srow[0];
  float acc = 0.f;
  for (int e = 0; e < nthis; ++e) {
    const int r = srow[e];
    if (r != cur) {
      atomicAdd(&out[cur * DFEAT + t], acc);
      acc = 0.f;
      cur = r;
    }
    acc = fmaf(pre[scol[e] * DFEAT + t], sval[e], acc);
  }
  atomicAdd(&out[cur * DFEAT + t], acc);
}

// ---------------------------------------------------------------------------
// Host launch
// Inputs: x[N*128] f32, W[128*128] f32, row[E] i32, col[E] i32, vals[E] f32
// Output: out[N*128] f32. Workspace: pre[N*128] f32 (51.2 MB) in d_ws.
// ---------------------------------------------------------------------------
extern "C" void kernel_launch(void* const* d_in, const int* in_sizes, int n_in,
                              void* d_out, int out_size, void* d_ws,
                              size_t ws_size, hipStream_t stream) {
  const float* x = (const float*)d_in[0];
  const float* W = (const float*)d_in[1];
  const int* row = (const int*)d_in[2];
  const int* col = (const int*)d_in[3];
  const float* vals = (const float*)d_in[4];
  float* out = (float*)d_out;
  float* pre = (float*)d_ws;

  const int nrows = in_sizes[0] / DFEAT;  // 100000
  const int nE = in_sizes[2];             // 1600000
  const int n4 = (nrows * DFEAT) / 4;     // float4 count of out/pre

  // 1) zero-init accumulator output
  {
    int grid = (n4 + 255) / 256;
    if (grid > 8192) grid = 8192;
    zero_f32<<<grid, 256, 0, stream>>>(out, n4);
  }

  // 2) pre = x @ W  (WMMA f32)
  {
    int ntiles = nrows >> 4;  // 6250
    int grid = ntiles < 2048 ? ntiles : 2048;
    gemm_xw_wmma<<<grid, 256, 0, stream>>>(x, W, pre, nrows);
  }

  // 3) sparse scatter-add
  {
    int grid = (nE + SPMM_CHUNK - 1) / SPMM_CHUNK;  // 12500
    spmm_scatter<<<grid, SPMM_CHUNK, 0, stream>>>(pre, row, col, vals, out, nE);
  }

  // 4) ReLU
  {
    int grid = (n4 + 255) / 256;
    if (grid > 8192) grid = 8192;
    relu_f32<<<grid, 256, 0, stream>>>(out, n4);
  }
}